// yoloLoss_18949395710676
// MI455X (gfx1250) — compile-verified
//
#include <hip/hip_runtime.h>

// ---------------------------------------------------------------------------
// YOLO loss on MI455X (gfx1250): memory-bound streaming reduction.
//   pass 1: grid-stride over 802,816 cells, per-thread f32 accumulation,
//           wave32 reduction via V_WMMA_F32_16X16X4_F32, per-block partial
//           to d_ws (1024 floats).
//   pass 2: single block reduces the 1024 partials deterministically and
//           scales by 1/4096.
// ---------------------------------------------------------------------------

typedef float v2f __attribute__((ext_vector_type(2)));
typedef float v8f __attribute__((ext_vector_type(8)));

static constexpr int   kCells   = 4096 * 14 * 14;   // B*S*S
static constexpr int   kBlocks  = 1024;
static constexpr int   kThreads = 256;

// Wave32 sum using V_WMMA_F32_16X16X4_F32.
// A layout (ISA 7.12.2, 32-bit A 16x4): lane m (0-15) VGPR0 = A[m][0],
// VGPR1 = A[m][1]; lane m+16 VGPR0 = A[m][2], VGPR1 = A[m][3].
// With a = (x, 0) and B = all-ones:  D[m][n] = x_m + x_{m+16}.
// D layout: VGPR j holds row M=j (lanes 0-15) / M=j+8 (lanes 16-31), so
// summing the 8 D registers per lane gives half-sums; one shfl_xor(16)
// completes the full wave sum (exact f32, deterministic order).
__device__ __forceinline__ float wave_sum_wmma(float x) {
    v2f a; a[0] = x;    a[1] = 0.0f;
    v2f b; b[0] = 1.0f; b[1] = 1.0f;
    v8f c = {};
    v8f d = __builtin_amdgcn_wmma_f32_16x16x4_f32(
        /*neg_a=*/false, a, /*neg_b=*/false, b,
        /*c_mod=*/(short)0, c, /*reuse_a=*/false, /*reuse_b=*/false);
    float s = ((d[0] + d[1]) + (d[2] + d[3])) + ((d[4] + d[5]) + (d[6] + d[7]));
    s += __shfl_xor(s, 16, 32);
    return s;
}

// Per-cell loss. p2/t2 point at the cell's 30 floats viewed as 15 float2
// (cell stride 120 B => always 8-byte aligned => global_load_b64).
__device__ __forceinline__ float cell_loss(const float2* __restrict__ p2,
                                           const float2* __restrict__ t2) {
    // First 10 channels of each (boxes + confidences).
    float2 pa0 = p2[0], pa1 = p2[1], pa2 = p2[2], pa3 = p2[3], pa4 = p2[4];
    float2 ta0 = t2[0], ta1 = t2[1], ta2 = t2[2], ta4 = t2[4];

    // Class loss accumulated on the fly over channels 10..29.
    float cls = 0.0f;
#pragma unroll
    for (int k = 5; k < 15; ++k) {
        float2 pc = p2[k], tc = t2[k];
        float dx = pc.x - tc.x;
        float dy = pc.y - tc.y;
        cls = fmaf(dx, dx, cls);
        cls = fmaf(dy, dy, cls);
    }

    bool obj = ta2.x > 0.0f;                 // target[...,4]

    // No-object confidence loss (both predicted confidences).
    float dn0 = pa2.x - ta2.x;               // pred[4] - target[4]
    float dn1 = pa4.y - ta4.y;               // pred[9] - target[9]
    float noobj = dn0 * dn0 + dn1 * dn1;

    // Target box extents.
    float tcx = ta0.x / 14.0f, tcy = ta0.y / 14.0f;
    float tw = ta1.x, th = ta1.y;
    float tminx = tcx - 0.5f * tw, tmaxx = tcx + 0.5f * tw;
    float tminy = tcy - 0.5f * th, tmaxy = tcy + 0.5f * th;
    float area_t = (tmaxx - tminx) * (tmaxy - tminy);

    // Box 0: pred[0..3], conf pred[4].
    float c0x = pa0.x / 14.0f, c0y = pa0.y / 14.0f;
    float w0 = pa1.x, h0 = pa1.y;
    float b0minx = c0x - 0.5f * w0, b0maxx = c0x + 0.5f * w0;
    float b0miny = c0y - 0.5f * h0, b0maxy = c0y + 0.5f * h0;
    float iw0 = fmaxf(fminf(b0maxx, tmaxx) - fmaxf(b0minx, tminx), 0.0f);
    float ih0 = fmaxf(fminf(b0maxy, tmaxy) - fmaxf(b0miny, tminy), 0.0f);
    float inter0 = iw0 * ih0;
    float area0 = (b0maxx - b0minx) * (b0maxy - b0miny);
    float iou0 = inter0 / (area0 + area_t - inter0);

    // Box 1: pred[5..8], conf pred[9].
    float c1x = pa2.y / 14.0f, c1y = pa3.x / 14.0f;
    float w1 = pa3.y, h1 = pa4.x;
    float b1minx = c1x - 0.5f * w1, b1maxx = c1x + 0.5f * w1;
    float b1miny = c1y - 0.5f * h1, b1maxy = c1y + 0.5f * h1;
    float iw1 = fmaxf(fminf(b1maxx, tmaxx) - fmaxf(b1minx, tminx), 0.0f);
    float ih1 = fmaxf(fminf(b1maxy, tmaxy) - fmaxf(b1miny, tminy), 0.0f);
    float inter1 = iw1 * ih1;
    float area1 = (b1maxx - b1minx) * (b1maxy - b1miny);
    float iou1 = inter1 / (area1 + area_t - inter1);

    // argmax tie-break: box 0 wins ties (JAX first-occurrence semantics).
    bool r1 = iou1 > iou0;
    float max_iou = r1 ? iou1 : iou0;
    float rx = r1 ? pa2.y : pa0.x;
    float ry = r1 ? pa3.x : pa0.y;
    float rw = r1 ? pa3.y : pa1.x;
    float rh = r1 ? pa4.x : pa1.y;
    float rc = r1 ? pa4.y : pa2.x;           // responsible confidence
    float nc = r1 ? pa2.x : pa4.y;           // non-responsible confidence

    float dcx = rx - ta0.x;
    float dcy = ry - ta0.y;
    float dsw = sqrtf(rw) - sqrtf(ta1.x);
    float dsh = sqrtf(rh) - sqrtf(ta1.y);
    float loc = (dcx * dcx + dcy * dcy) + (dsw * dsw + dsh * dsh);
    float dcf = rc - max_iou;

    float objloss = 5.0f * loc + 2.0f * (dcf * dcf) + nc * nc + cls;
    return obj ? objloss : 0.5f * noobj;
}

__global__ __launch_bounds__(kThreads)
void yolo_partial(const float* __restrict__ pred,
                  const float* __restrict__ target,
                  float* __restrict__ ws) {
    const int tid    = blockIdx.x * blockDim.x + threadIdx.x;
    const int stride = gridDim.x * blockDim.x;

    float acc = 0.0f;
    for (int c = tid; c < kCells; c += stride) {
        // Prefetch the next cell this thread will touch (global_prefetch_b8).
        int cn = c + stride;
        if (cn < kCells) {
            __builtin_prefetch(pred + (size_t)cn * 30, 0, 0);
            __builtin_prefetch(target + (size_t)cn * 30, 0, 0);
        }
        const float2* p2 = reinterpret_cast<const float2*>(pred)   + (size_t)c * 15;
        const float2* t2 = reinterpret_cast<const float2*>(target) + (size_t)c * 15;
        acc += cell_loss(p2, t2);
    }

    // Uniform control flow from here: EXEC is all-1s (WMMA requirement).
    float wsum = wave_sum_wmma(acc);

    __shared__ float smem[kThreads / 32];
    const int lane = threadIdx.x & 31;
    const int wid  = threadIdx.x >> 5;
    if (lane == 0) smem[wid] = wsum;
    __syncthreads();
    if (threadIdx.x == 0) {
        float s = 0.0f;
#pragma unroll
        for (int i = 0; i < kThreads / 32; ++i) s += smem[i];
        ws[blockIdx.x] = s;
    }
}

__global__ __launch_bounds__(kThreads)
void yolo_finalize(const float* __restrict__ ws, float* __restrict__ out) {
    float acc = 0.0f;
    // kBlocks % kThreads == 0 -> uniform trip count, EXEC stays all-1s.
    for (int i = threadIdx.x; i < kBlocks; i += kThreads) acc += ws[i];

    float wsum = wave_sum_wmma(acc);

    __shared__ float smem[kThreads / 32];
    const int lane = threadIdx.x & 31;
    const int wid  = threadIdx.x >> 5;
    if (lane == 0) smem[wid] = wsum;
    __syncthreads();
    if (threadIdx.x == 0) {
        float s = 0.0f;
#pragma unroll
        for (int i = 0; i < kThreads / 32; ++i) s += smem[i];
        out[0] = s * (1.0f / 4096.0f);       // exact: divide by n = B = 2^12
    }
}

extern "C" void kernel_launch(void* const* d_in, const int* in_sizes, int n_in,
                              void* d_out, int out_size, void* d_ws, size_t ws_size,
                              hipStream_t stream) {
    const float* pred   = (const float*)d_in[0];
    const float* target = (const float*)d_in[1];
    float* out = (float*)d_out;
    float* ws  = (float*)d_ws;   // needs kBlocks * 4 = 4 KB of scratch

    yolo_partial<<<kBlocks, kThreads, 0, stream>>>(pred, target, ws);
    yolo_finalize<<<1, kThreads, 0, stream>>>(ws, out);
}